// UncertaintyModel_11312943857710
// MI455X (gfx1250) — compile-verified
//
#include <hip/hip_runtime.h>

typedef __attribute__((ext_vector_type(2))) float v2f;
typedef __attribute__((ext_vector_type(8))) float v8f;

#define B_  16
#define D_  128
#define H_  2048
#define O_  4
#define W_  (O_ * H_ + O_)   // 8196
#define OH_ ((size_t)O_ * H_)  // 8192

static __device__ __forceinline__ v8f wmma_f32(v2f a, v2f b, v8f c) {
  return __builtin_amdgcn_wmma_f32_16x16x4_f32(false, a, false, b,
                                               (short)0, c, false, false);
}

// ---------------------------------------------------------------------------
// Kernel A: h = relu(x @ W1^T + b1)
// One wave per 16-wide j-tile; M=16 batch rows, K=128 in steps of 4.
// ---------------------------------------------------------------------------
__global__ __launch_bounds__(32) void k_hidden(const float* __restrict__ x,
                                               const float* __restrict__ W1,
                                               const float* __restrict__ b1,
                                               float* __restrict__ h) {
  const int lane  = threadIdx.x;
  const int half  = lane >> 4;
  const int l16   = lane & 15;
  const int nbase = blockIdx.x * 16;

  const float* xrow  = x  + (size_t)l16 * D_;            // A row m = l16
  const float* w1row = W1 + (size_t)(nbase + l16) * D_;  // B col n = l16

  v8f acc = {};
  for (int k0 = 0; k0 < D_; k0 += 4) {
    const int k = k0 + 2 * half;
    v2f a, b;
    a.x = xrow[k];   a.y = xrow[k + 1];
    b.x = w1row[k];  b.y = w1row[k + 1];
    acc = wmma_f32(a, b, acc);
  }

  const float bias = b1[nbase + l16];
  #pragma unroll
  for (int r = 0; r < 8; ++r) {
    const int m = r + 8 * half;            // C/D: lanes 0-15 -> M=r, 16-31 -> M=r+8
    float v = acc[r] + bias;
    h[(size_t)m * H_ + nbase + l16] = v > 0.0f ? v : 0.0f;
  }
}

// ---------------------------------------------------------------------------
// Kernel init: plain stores of the bias / constant terms (deterministic base
// that later kernels accumulate onto with atomics).
//   pred[b,o] = b2[o]
//   unc[b,o]  = C[OH+o, OH+o]
// ---------------------------------------------------------------------------
__global__ __launch_bounds__(64) void k_init(const float* __restrict__ b2,
                                             const float* __restrict__ C,
                                             float* __restrict__ pred,
                                             float* __restrict__ unc) {
  const int t = threadIdx.x;            // 0..63 -> (b,o)
  const int o = t & 3;
  pred[t] = b2[o];
  unc[t]  = C[(OH_ + o) * (size_t)W_ + OH_ + o];
}

// ---------------------------------------------------------------------------
// Kernel lin: predictions dot-products + linear uncertainty terms.
//   pred[b,o] += sum_i h[b,i] * W2[o,i]
//   unc[b,o]  += sum_i h[b,i] * (C[oH+i, OH+o] + C[OH+o, oH+i])
// Block = (ix=16, b=16); grid = (o=4, chunk=8); each thread sums 16 i-values,
// then a 16-lane shfl_xor reduction and one atomicAdd per (b,o) per block.
// ---------------------------------------------------------------------------
__global__ __launch_bounds__(256) void k_lin(const float* __restrict__ h,
                                             const float* __restrict__ W2,
                                             const float* __restrict__ C,
                                             float* __restrict__ pred,
                                             float* __restrict__ unc) {
  const int ix = threadIdx.x;           // 0..15 (fastest -> 16-lane halves)
  const int b  = threadIdx.y;           // 0..15
  const int o  = blockIdx.x;            // 0..3
  const int c0 = blockIdx.y * 256;      // i-chunk base (8 chunks of 256)
  const size_t Wd = (size_t)W_;
  const size_t oH = (size_t)o * H_;

  const float* hrow  = h  + (size_t)b * H_;
  const float* w2row = W2 + (size_t)o * H_;
  const float* colp  = C + oH * Wd + OH_ + o;     // C[oH+i, OH+o], stride Wd
  const float* rowp  = C + (OH_ + o) * Wd + oH;   // C[OH+o, oH+i], stride 1

  float p = 0.0f, u = 0.0f;
  #pragma unroll
  for (int s = 0; s < 16; ++s) {
    const int i = c0 + ix + 16 * s;
    const float hv = hrow[i];
    p += hv * w2row[i];
    u += hv * (colp[(size_t)i * Wd] + rowp[i]);
  }
  // reduce over ix: xor masks < 16 stay within each 16-lane half-wave
  #pragma unroll
  for (int mask = 1; mask < 16; mask <<= 1) {
    p += __shfl_xor(p, mask, 32);
    u += __shfl_xor(u, mask, 32);
  }
  if (ix == 0) {
    atomicAdd(&pred[b * O_ + o], p);
    atomicAdd(&unc[b * O_ + o], u);
  }
}

// ---------------------------------------------------------------------------
// Kernel quad: u[b,o] += h_b . M_oo . h_b   (M_oo = inv_cov diag block o)
// Grid (128 n-tiles, 4 heads); block = 4 waves, each owning a 512-wide
// K-chunk. Two alternating WMMA accumulators break the RAW chain. Per-wave
// 16-lane shfl reduction over N, cross-wave combine in LDS, 16 atomics/block.
// ---------------------------------------------------------------------------
__global__ __launch_bounds__(128) void k_quad(const float* __restrict__ h,
                                              const float* __restrict__ C,
                                              float* __restrict__ unc) {
  const int tid   = threadIdx.x;
  const int kw    = tid >> 5;          // wave id = k-chunk 0..3
  const int lane  = tid & 31;
  const int half  = lane >> 4;
  const int l16   = lane & 15;
  const int nbase = blockIdx.x * 16;
  const int o     = blockIdx.y;
  const size_t Wd = (size_t)W_;
  const size_t oH = (size_t)o * H_;
  const int kbase = kw * (H_ / 4);     // 512-wide K chunk per wave

  const float* hA    = h + (size_t)l16 * H_;              // A row m = l16
  const float* Bbase = C + oH * Wd + oH + nbase + l16;    // M_oo[k][n=l16]

  v8f acc0 = {}, acc1 = {};
  for (int k0 = kbase; k0 < kbase + H_ / 4; k0 += 8) {
    {
      const int k = k0 + 2 * half;
      v2f a, b;
      a.x = hA[k];  a.y = hA[k + 1];
      b.x = Bbase[(size_t)k * Wd];
      b.y = Bbase[(size_t)(k + 1) * Wd];
      acc0 = wmma_f32(a, b, acc0);
    }
    {
      const int k = k0 + 4 + 2 * half;
      v2f a, b;
      a.x = hA[k];  a.y = hA[k + 1];
      b.x = Bbase[(size_t)k * Wd];
      b.y = Bbase[(size_t)(k + 1) * Wd];
      acc1 = wmma_f32(a, b, acc1);
    }
  }

  // s[r] = G[m, nbase+l16] * h[m, nbase+l16],  m = r + 8*half
  float s[8];
  #pragma unroll
  for (int r = 0; r < 8; ++r) {
    const int m = r + 8 * half;
    s[r] = (acc0[r] + acc1[r]) * h[(size_t)m * H_ + nbase + l16];
  }
  #pragma unroll
  for (int mask = 1; mask < 16; mask <<= 1) {
    #pragma unroll
    for (int r = 0; r < 8; ++r)
      s[r] += __shfl_xor(s[r], mask, 32);
  }

  __shared__ float sh[4][16];
  if (l16 == 0) {                       // lanes 0 (m=0..7) and 16 (m=8..15)
    #pragma unroll
    for (int r = 0; r < 8; ++r)
      sh[kw][r + 8 * half] = s[r];
  }
  __syncthreads();
  if (tid < 16) {
    const float v = sh[0][tid] + sh[1][tid] + sh[2][tid] + sh[3][tid];
    atomicAdd(&unc[tid * O_ + o], v);
  }
}

extern "C" void kernel_launch(void* const* d_in, const int* in_sizes, int n_in,
                              void* d_out, int out_size, void* d_ws, size_t ws_size,
                              hipStream_t stream) {
  const float* x  = (const float*)d_in[0];
  const float* W1 = (const float*)d_in[1];
  const float* b1 = (const float*)d_in[2];
  const float* W2 = (const float*)d_in[3];
  const float* b2 = (const float*)d_in[4];
  const float* C  = (const float*)d_in[5];

  float* pred = (float*)d_out;             // 16*4 predictions
  float* unc  = (float*)d_out + B_ * O_;   // 16*4 uncertainty
  float* h    = (float*)d_ws;              // 16*2048 f32 = 128 KB scratch

  // h = relu(x @ W1^T + b1)
  k_hidden<<<dim3(H_ / 16), dim3(32), 0, stream>>>(x, W1, b1, h);

  // deterministic base values (plain stores each call)
  k_init<<<dim3(1), dim3(64), 0, stream>>>(b2, C, pred, unc);

  // predictions + linear uncertainty terms (parallel over i)
  k_lin<<<dim3(O_, 8), dim3(16, 16), 0, stream>>>(h, W2, C, pred, unc);

  // quadratic term: 512 blocks x 4 waves (K split), WMMA f32 streaming
  k_quad<<<dim3(H_ / 16, O_), dim3(128), 0, stream>>>(h, C, unc);
}